// XiLoss_90606630076637
// MI455X (gfx1250) — compile-verified
//
#include <hip/hip_runtime.h>
#include <hip/hip_bf16.h>
#include <stdint.h>

// ---------------------------------------------------------------------------
// Soft-rank Chatterjee-Xi loss for MI455X (gfx1250, wave32).
// Pipeline: bitonic argsort(y_true) -> gather/scale -> bitonic sort desc ->
// single-thread PAVA -> scatter ranks -> deterministic block reductions ->
// WMMA (f32 16x16x4) final reduction.
// CDNA5 specifics: v_wmma_f32_16x16x4_f32 reduction, async global->LDS tile
// staging (ASYNCcnt) in the fused LDS bitonic stages, global_prefetch_b8.
// ---------------------------------------------------------------------------

typedef __attribute__((ext_vector_type(2))) float v2f;
typedef __attribute__((ext_vector_type(8))) float v8f;

#define TAU_F      0.1f
#define LAMBDA_F   1.0f
#define NPART      512
#define THREADS    256
#define TILE       256

// ---- async global -> LDS copy of one b32 per lane (gfx1250, ASYNCcnt) -----
// VDST VGPR holds the LDS byte offset (flat shared address truncated to 32b,
// per ISA 10.2 LDS aperture mapping); VADDR is the 64-bit global address (GV).
__device__ __forceinline__ void async_ld_b32(void* lds_dst, const void* gsrc) {
    uint32_t loff = (uint32_t)(uintptr_t)lds_dst;
    uint64_t gaddr = (uint64_t)(uintptr_t)gsrc;
    asm volatile("global_load_async_to_lds_b32 %0, %1, off"
                 :: "v"(loff), "v"(gaddr)
                 : "memory");
}
__device__ __forceinline__ void wait_asynccnt0() {
    asm volatile("s_wait_asynccnt 0" ::: "memory");
}

// ---- init: key = y_true, val = identity -----------------------------------
__global__ void xi_init_kernel(float* __restrict__ key, int* __restrict__ val,
                               const float* __restrict__ y_true, int n) {
    int i = blockIdx.x * blockDim.x + threadIdx.x;
    if (i < n) {
        key[i] = y_true[i];
        val[i] = i;
    }
}

// ---- one global bitonic compare-exchange pass (j >= 256) ------------------
__global__ void xi_bitonic_kernel(float* __restrict__ key, int* __restrict__ val,
                                  int j, int k, int n, int descending) {
    unsigned i = blockIdx.x * blockDim.x + threadIdx.x;
    unsigned l = i ^ (unsigned)j;
    if (l > i && l < (unsigned)n) {
        float ki = key[i];
        float kl = key[l];
        bool up = ((i & (unsigned)k) == 0u);
        if (descending) up = !up;
        bool do_swap = up ? (ki > kl) : (ki < kl);
        if (do_swap) {
            key[i] = kl; key[l] = ki;
            int t = val[i]; val[i] = val[l]; val[l] = t;
        }
    }
}

// ---- fused LDS bitonic: all stages k=2..TILE (partner always in-tile) -----
__global__ void xi_sort_local_kernel(float* __restrict__ key,
                                     int* __restrict__ val,
                                     int n, int descending) {
    __shared__ float sk[TILE];
    __shared__ int   sv[TILE];
    int tid  = threadIdx.x;
    int base = blockIdx.x * TILE;
    int gi   = base + tid;

    async_ld_b32(&sk[tid], &key[gi]);       // -> global_load_async_to_lds_b32
    async_ld_b32(&sv[tid], &val[gi]);
    wait_asynccnt0();                        // -> s_wait_asynccnt 0
    __syncthreads();

    for (int k = 2; k <= TILE; k <<= 1) {
        for (int j = k >> 1; j > 0; j >>= 1) {
            if (tid < TILE / 2) {
                int i = ((tid & ~(j - 1)) << 1) | (tid & (j - 1));
                int p = i | j;
                bool up = (((base + i) & k) == 0);
                if (descending) up = !up;
                float ki = sk[i], kp = sk[p];
                bool sw = up ? (ki > kp) : (ki < kp);
                if (sw) {
                    sk[i] = kp; sk[p] = ki;
                    int t = sv[i]; sv[i] = sv[p]; sv[p] = t;
                }
            }
            __syncthreads();
        }
    }
    key[gi] = sk[tid];
    val[gi] = sv[tid];
}

// ---- fused LDS bitonic tail: fixed k, passes j=128..1 ---------------------
__global__ void xi_sort_tail_kernel(float* __restrict__ key,
                                    int* __restrict__ val,
                                    int k, int n, int descending) {
    __shared__ float sk[TILE];
    __shared__ int   sv[TILE];
    int tid  = threadIdx.x;
    int base = blockIdx.x * TILE;
    int gi   = base + tid;

    async_ld_b32(&sk[tid], &key[gi]);
    async_ld_b32(&sv[tid], &val[gi]);
    wait_asynccnt0();
    __syncthreads();

    for (int j = TILE / 2; j > 0; j >>= 1) {
        if (tid < TILE / 2) {
            int i = ((tid & ~(j - 1)) << 1) | (tid & (j - 1));
            int p = i | j;
            bool up = (((base + i) & k) == 0);
            if (descending) up = !up;
            float ki = sk[i], kp = sk[p];
            bool sw = up ? (ki > kp) : (ki < kp);
            if (sw) {
                sk[i] = kp; sk[p] = ki;
                int t = sv[i]; sv[i] = sv[p]; sv[p] = t;
            }
        }
        __syncthreads();
    }
    key[gi] = sk[tid];
    val[gi] = sv[tid];
}

// ---- gather permuted y_pred, scale by 1/tau, reset val=i ------------------
__global__ void xi_gather_kernel(float* __restrict__ key, int* __restrict__ val,
                                 const float* __restrict__ y_pred,
                                 float inv_tau, int n) {
    int i = blockIdx.x * blockDim.x + threadIdx.x;
    if (i < n) {
        int p = val[i];                 // perm from sort #1 (ascending y_true)
        key[i] = y_pred[p] * inv_tau;   // z in permuted order
        val[i] = i;                     // index within permuted order
    }
}

// ---- PAVA (nonincreasing isotonic L2), strictly sequential ----------------
__global__ void xi_pava_kernel(const float* __restrict__ skey,
                               float* __restrict__ sol,
                               float* __restrict__ sums,
                               float* __restrict__ cnts,
                               int* __restrict__ ends, int n) {
    if (blockIdx.x != 0 || threadIdx.x != 0) return;
    int top = -1;
    for (int i = 0; i < n; ++i) {
        float s = skey[i] - (float)(n - i);
        float c = 1.0f;
        while (top >= 0 && s * cnts[top] >= sums[top] * c) {
            s += sums[top];
            c += cnts[top];
            --top;
        }
        ++top;
        sums[top] = s;
        cnts[top] = c;
        ends[top] = i;
    }
    int start = 0;
    for (int b = 0; b <= top; ++b) {
        float mean = sums[b] / cnts[b];
        int e = ends[b];
        for (int jj = start; jj <= e; ++jj) sol[jj] = mean;
        start = e + 1;
    }
}

// ---- ranks[perm2[i]] = s[i] - sol[i] --------------------------------------
__global__ void xi_scatter_kernel(float* __restrict__ ranks,
                                  const float* __restrict__ skey,
                                  const float* __restrict__ sol,
                                  const int* __restrict__ val, int n) {
    int i = blockIdx.x * blockDim.x + threadIdx.x;
    if (i < n) {
        ranks[val[i]] = skey[i] - sol[i];
    }
}

// ---- deterministic block reduction: sum |ranks[i+1]-ranks[i]| -------------
__global__ void xi_absdiff_kernel(const float* __restrict__ r,
                                  float* __restrict__ part, int n) {
    __shared__ float sh[THREADS];
    int tid = threadIdx.x;
    float acc = 0.0f;
    for (int i = blockIdx.x * THREADS + tid; i < n - 1; i += gridDim.x * THREADS) {
        acc += fabsf(r[i + 1] - r[i]);
    }
    sh[tid] = acc;
    __syncthreads();
    for (int s = THREADS / 2; s > 0; s >>= 1) {
        if (tid < s) sh[tid] += sh[tid + s];
        __syncthreads();
    }
    if (tid == 0) part[blockIdx.x] = sh[0];
}

// ---- deterministic block reduction: sum (yp-yt)^2 -------------------------
__global__ void xi_mse_kernel(const float* __restrict__ yp,
                              const float* __restrict__ yt,
                              float* __restrict__ part, int n) {
    __shared__ float sh[THREADS];
    int tid = threadIdx.x;
    float acc = 0.0f;
    for (int i = blockIdx.x * THREADS + tid; i < n; i += gridDim.x * THREADS) {
        __builtin_prefetch(yp + i + 8192, 0, 0);   // -> global_prefetch_b8
        __builtin_prefetch(yt + i + 8192, 0, 0);
        float d = yp[i] - yt[i];
        acc += d * d;
    }
    sh[tid] = acc;
    __syncthreads();
    for (int s = THREADS / 2; s > 0; s >>= 1) {
        if (tid < s) sh[tid] += sh[tid + s];
        __syncthreads();
    }
    if (tid == 0) part[blockIdx.x] = sh[0];
}

// ---- final: WMMA f32 reduction of the two 512-entry partial arrays --------
// A = all-ones 16x4, B = 64 partials => D[m][n] = C[m][n] + colsum(B,n).
// Summing all 16 columns of any row of D yields the exact total of every B
// element, independent of the hardware (lane -> K,N) layout of B.
__global__ void xi_final_kernel(const float* __restrict__ partA,
                                const float* __restrict__ partB,
                                float* __restrict__ out, int npart, int n) {
    int lane = threadIdx.x;              // 0..31, single wave32
    v2f ones = {1.0f, 1.0f};
    v8f cA = {};
    v8f cB = {};
    for (int base = 0; base < npart; base += 64) {
        v2f bA, bB;
        bA[0] = partA[base + lane];
        bA[1] = partA[base + 32 + lane];
        bB[0] = partB[base + lane];
        bB[1] = partB[base + 32 + lane];
        cA = __builtin_amdgcn_wmma_f32_16x16x4_f32(
                 false, ones, false, bA, (short)0, cA, false, false);
        cB = __builtin_amdgcn_wmma_f32_16x16x4_f32(
                 false, ones, false, bB, (short)0, cB, false, false);
    }
    float dsum = cA[0];
    float msum = cB[0];
    for (int off = 8; off >= 1; off >>= 1) {
        dsum += __shfl_xor(dsum, off, 16);
        msum += __shfl_xor(msum, off, 16);
    }
    if (lane == 0) {
        float nf    = (float)n;
        float diffs = dsum;
        float xi    = 1.0f - 3.0f * diffs / (nf * nf - 1.0f);
        float task  = msum / nf;
        out[0] = task - LAMBDA_F * xi;   // total
        out[1] = xi;                     // xi_soft
    }
}

// ---------------------------------------------------------------------------
extern "C" void kernel_launch(void* const* d_in, const int* in_sizes, int n_in,
                              void* d_out, int out_size, void* d_ws, size_t ws_size,
                              hipStream_t stream) {
    const float* y_pred = (const float*)d_in[0];
    const float* y_true = (const float*)d_in[1];
    const int n = in_sizes[0];           // 131072, power of two

    // Workspace carve-up (7*N floats + 2*NPART floats ~ 3.7 MB)
    char* w = (char*)d_ws;
    float* key  = (float*)w; w += (size_t)n * sizeof(float);
    int*   val  = (int*)  w; w += (size_t)n * sizeof(int);
    float* sol  = (float*)w; w += (size_t)n * sizeof(float);
    float* rnk  = (float*)w; w += (size_t)n * sizeof(float);
    float* sums = (float*)w; w += (size_t)n * sizeof(float);
    float* cnts = (float*)w; w += (size_t)n * sizeof(float);
    int*   ends = (int*)  w; w += (size_t)n * sizeof(int);
    float* pA   = (float*)w; w += (size_t)NPART * sizeof(float);
    float* pB   = (float*)w;

    const int blocks = (n + THREADS - 1) / THREADS;

    // ---- Sort #1: ascending argsort of y_true -> val = perm --------------
    xi_init_kernel<<<blocks, THREADS, 0, stream>>>(key, val, y_true, n);
    xi_sort_local_kernel<<<blocks, THREADS, 0, stream>>>(key, val, n, 0);
    for (int k = TILE * 2; k <= n; k <<= 1) {
        for (int j = k >> 1; j >= TILE; j >>= 1)
            xi_bitonic_kernel<<<blocks, THREADS, 0, stream>>>(key, val, j, k, n, 0);
        xi_sort_tail_kernel<<<blocks, THREADS, 0, stream>>>(key, val, k, n, 0);
    }

    // ---- z = y_pred[perm] / tau; val = position index --------------------
    xi_gather_kernel<<<blocks, THREADS, 0, stream>>>(key, val, y_pred,
                                                     1.0f / TAU_F, n);

    // ---- Sort #2: descending -> key = s (desc), val = perm2 --------------
    xi_sort_local_kernel<<<blocks, THREADS, 0, stream>>>(key, val, n, 1);
    for (int k = TILE * 2; k <= n; k <<= 1) {
        for (int j = k >> 1; j >= TILE; j >>= 1)
            xi_bitonic_kernel<<<blocks, THREADS, 0, stream>>>(key, val, j, k, n, 1);
        xi_sort_tail_kernel<<<blocks, THREADS, 0, stream>>>(key, val, k, n, 1);
    }

    // ---- PAVA + expansion (sequential by nature) -------------------------
    xi_pava_kernel<<<1, 1, 0, stream>>>(key, sol, sums, cnts, ends, n);

    // ---- ranks[perm2[i]] = s[i] - sol[i] ---------------------------------
    xi_scatter_kernel<<<blocks, THREADS, 0, stream>>>(rnk, key, sol, val, n);

    // ---- Partial reductions (deterministic) ------------------------------
    xi_absdiff_kernel<<<NPART, THREADS, 0, stream>>>(rnk, pA, n);
    xi_mse_kernel<<<NPART, THREADS, 0, stream>>>(y_pred, y_true, pB, n);

    // ---- Final combine via f32 WMMA reduction (single wave) --------------
    xi_final_kernel<<<1, 32, 0, stream>>>(pA, pB, (float*)d_out, NPART, n);
}